// DetectionLoss_11072425689903
// MI455X (gfx1250) — compile-verified
//
#include <hip/hip_runtime.h>
#include <stdint.h>

#define LAMBDA_BOX  7.5f
#define LAMBDA_CONF 1.0f
#define IOU_THRESH  0.3f
#define F_ALPHA     0.25f
#define F_EPS       1e-7f

static constexpr int kB  = 64;     // batch
static constexpr int kN  = 8400;   // preds per image
static constexpr int kM  = 64;     // targets per image
static constexpr int TPB = 256;    // threads per block (8 waves on wave32)
static constexpr int BPI = (kN + TPB - 1) / TPB;  // 33 blocks per image

// ---------- gfx1250 async global->LDS helpers ----------
typedef int v4i __attribute__((vector_size(16)));
typedef __attribute__((address_space(1))) v4i* glb_v4i_p;
typedef __attribute__((address_space(3))) v4i* lds_v4i_p;

#if defined(__has_builtin)
#if __has_builtin(__builtin_amdgcn_global_load_async_to_lds_b128)
#define HAVE_ASYNC_BUILTIN 1
#endif
#if __has_builtin(__builtin_amdgcn_s_wait_asynccnt)
#define HAVE_ASYNCCNT_BUILTIN 1
#endif
#endif

__device__ __forceinline__ void async_copy_b128(void* lds_dst, const void* gsrc) {
#if defined(HAVE_ASYNC_BUILTIN)
  // (global src, lds dst, imm offset, imm cpol)
  __builtin_amdgcn_global_load_async_to_lds_b128(
      (glb_v4i_p)(uintptr_t)gsrc,
      (lds_v4i_p)(uint32_t)(uintptr_t)lds_dst, 0, 0);
#else
  asm volatile("global_load_async_to_lds_b128 %0, %1, off"
               :: "v"((uint32_t)(uintptr_t)lds_dst), "v"(gsrc)
               : "memory");
#endif
}

__device__ __forceinline__ void wait_async_zero() {
#if defined(HAVE_ASYNCCNT_BUILTIN)
  __builtin_amdgcn_s_wait_asynccnt(0);
#else
  asm volatile("s_wait_asynccnt 0x0" ::: "memory");
#endif
}

// ---------- deterministic wave32 reduction ----------
__device__ __forceinline__ float wave_sum32(float v) {
#pragma unroll
  for (int off = 16; off > 0; off >>= 1)
    v += __shfl_down(v, off, 32);
  return v;
}

// ============================================================
// Kernel 1: per-(image, chunk-of-256-preds) partial sums.
// partials[(b*BPI + blk)*3 + {0:conf_sum, 1:box_sum, 2:npos}]
// ============================================================
__global__ __launch_bounds__(TPB) void det_loss_main(
    const float* __restrict__ preds,     // [B, N, 5] cxcywh + logit
    const float* __restrict__ targets,   // [B, M, 4] cxcywh
    float* __restrict__ partials) {
  __shared__ __align__(16) float pred_stage[TPB * 5];  // staged pred chunk
  __shared__ __align__(16) float tgt_raw[kM * 4];      // raw cxcywh targets
  __shared__ float sx1[kM], sy1[kM], sx2[kM], sy2[kM], sar[kM];  // SoA xyxy
  __shared__ float wred[TPB / 32][3];

  const int b   = blockIdx.y;
  const int blk = blockIdx.x;
  const int tid = threadIdx.x;

  const float* tgt_g = targets + (size_t)b * kM * 4;
  const int n0  = blk * TPB;
  const int cnt = min(TPB, kN - n0);
  const float* pred_g = preds + ((size_t)b * kN + n0) * 5;

  __builtin_prefetch(tgt_g, 0, 0);  // global_prefetch_b8

  // ---- async-stage targets (1 KB) and this block's pred chunk into LDS ----
  if (tid < kM)
    async_copy_b128(&tgt_raw[tid * 4], tgt_g + tid * 4);
  const int nvec = (cnt * 5) >> 2;  // cnt*5 floats, 16B granules (divides evenly)
  for (int i = tid; i < nvec; i += TPB)
    async_copy_b128(&pred_stage[i * 4], pred_g + i * 4);
  wait_async_zero();
  __syncthreads();

  // ---- convert targets cxcywh -> xyxy SoA (+area), mirroring reference ----
  if (tid < kM) {
    float cx = tgt_raw[tid * 4 + 0], cy = tgt_raw[tid * 4 + 1];
    float w  = tgt_raw[tid * 4 + 2], h  = tgt_raw[tid * 4 + 3];
    float x1 = cx - 0.5f * w, y1 = cy - 0.5f * h;
    float x2 = cx + 0.5f * w, y2 = cy + 0.5f * h;
    sx1[tid] = x1; sy1[tid] = y1; sx2[tid] = x2; sy2[tid] = y2;
    sar[tid] = (x2 - x1) * (y2 - y1);
  }
  // my prediction (distinct LDS region from the SoA writes above)
  float pcx = pred_stage[tid * 5 + 0];
  float pcy = pred_stage[tid * 5 + 1];
  float pw0 = pred_stage[tid * 5 + 2];
  float ph0 = pred_stage[tid * 5 + 3];
  float z   = pred_stage[tid * 5 + 4];
  __syncthreads();

  float conf_acc = 0.f, box_acc = 0.f, pos_acc = 0.f;
  if (tid < cnt) {
    float px1 = pcx - 0.5f * pw0, py1 = pcy - 0.5f * ph0;
    float px2 = pcx + 0.5f * pw0, py2 = pcy + 0.5f * ph0;
    float pa  = (px2 - px1) * (py2 - py1);

    // ---- max/argmax IoU over 64 targets (LDS broadcast reads) ----
    float best = -1.f;
    int bm = 0;
#pragma unroll 8
    for (int m = 0; m < kM; ++m) {
      float iw = fminf(px2, sx2[m]) - fmaxf(px1, sx1[m]);
      float ih = fminf(py2, sy2[m]) - fmaxf(py1, sy1[m]);
      iw = fmaxf(iw, 0.f);
      ih = fmaxf(ih, 0.f);
      float inter = iw * ih;
      float iou = inter / (pa + sar[m] - inter);  // reference: no EPS here
      if (iou > best) { best = iou; bm = m; }     // first-max (argmax semantics)
    }
    const bool matched = best > IOU_THRESH;

    // ---- focal BCE with logits (numerically stable, as reference) ----
    float t   = matched ? 1.f : 0.f;
    float bce = fmaxf(z, 0.f) - z * t + log1pf(expf(-fabsf(z)));
    float pt  = expf(-bce);
    float om  = 1.f - pt;
    conf_acc  = F_ALPHA * om * om * bce;  // GAMMA = 2

    // ---- CIoU only for matched preds ----
    if (matched) {
      float gx1 = sx1[bm], gy1 = sy1[bm], gx2 = sx2[bm], gy2 = sy2[bm];
      float iw = fmaxf(fminf(px2, gx2) - fmaxf(px1, gx1), 0.f);
      float ih = fmaxf(fminf(py2, gy2) - fmaxf(py1, gy1), 0.f);
      float inter = iw * ih;
      float wp = px2 - px1, hp = py2 - py1;
      float wg = gx2 - gx1, hg = gy2 - gy1;
      float uni = wp * hp + wg * hg - inter;
      float iou = inter / (uni + F_EPS);
      float dxe = fmaxf(px2, gx2) - fminf(px1, gx1);
      float dye = fmaxf(py2, gy2) - fminf(py1, gy1);
      float diag = dxe * dxe + dye * dye + F_EPS;
      float cxd = (gx1 + gx2) * 0.5f - (px1 + px2) * 0.5f;
      float cyd = (gy1 + gy2) * 0.5f - (py1 + py2) * 0.5f;
      float dist = cxd * cxd + cyd * cyd;
      float dv = atanf(wg / hg) - atanf(wp / hp);
      float v  = 0.405284734569f * dv * dv;  // 4/pi^2
      float alpha = v / (1.f - iou + v + F_EPS);
      box_acc = (1.f - iou) + dist / diag + alpha * v;
      pos_acc = 1.f;
    }
  }

  // ---- deterministic block reduction: wave32 shuffles + LDS ----
  const int wave = tid >> 5, lane = tid & 31;
  float c = wave_sum32(conf_acc);
  float x = wave_sum32(box_acc);
  float p = wave_sum32(pos_acc);
  if (lane == 0) { wred[wave][0] = c; wred[wave][1] = x; wred[wave][2] = p; }
  __syncthreads();
  if (tid == 0) {
    float C = 0.f, X = 0.f, P = 0.f;
#pragma unroll
    for (int w = 0; w < TPB / 32; ++w) {
      C += wred[w][0]; X += wred[w][1]; P += wred[w][2];
    }
    float* out = partials + ((size_t)b * BPI + blk) * 3;
    out[0] = C; out[1] = X; out[2] = P;
  }
}

// ============================================================
// Kernel 2: reduce partials exactly in the reference's order.
// ============================================================
__global__ __launch_bounds__(64) void det_loss_final(
    const float* __restrict__ partials, float* __restrict__ out) {
  __shared__ float s[kB][3];
  const int b = threadIdx.x;
  if (b < kB) {
    float C = 0.f, X = 0.f, P = 0.f;
    for (int j = 0; j < BPI; ++j) {
      const float* p = partials + ((size_t)b * BPI + j) * 3;
      C += p[0]; X += p[1]; P += p[2];
    }
    s[b][0] = C; s[b][1] = X; s[b][2] = P;
  }
  __syncthreads();
  if (b == 0) {
    float conf_total = 0.f, box_total = 0.f, nimg = 0.f;
    for (int i = 0; i < kB; ++i) {
      conf_total += s[i][0] / (float)kN;     // per-image focal mean
      float np = s[i][2];
      if (np > 0.f) { box_total += s[i][1] / np; nimg += 1.f; }
    }
    float conf_loss = conf_total / (float)kB;
    float box_loss  = box_total / fmaxf(nimg, 1.f);
    out[0] = LAMBDA_CONF * conf_loss + LAMBDA_BOX * box_loss;
  }
}

extern "C" void kernel_launch(void* const* d_in, const int* in_sizes, int n_in,
                              void* d_out, int out_size, void* d_ws, size_t ws_size,
                              hipStream_t stream) {
  const float* preds   = (const float*)d_in[0];   // [64, 8400, 5] f32
  const float* targets = (const float*)d_in[1];   // [64, 64, 4] f32
  float* partials      = (float*)d_ws;            // 64*33*3 floats = ~25 KB

  dim3 grid(BPI, kB);
  det_loss_main<<<grid, TPB, 0, stream>>>(preds, targets, partials);
  det_loss_final<<<1, 64, 0, stream>>>(partials, (float*)d_out);
}